// AllTnn_11819749999087
// MI455X (gfx1250) — compile-verified
//
#include <hip/hip_runtime.h>
#include <hip/hip_bf16.h>

typedef __attribute__((ext_vector_type(2))) float v2f;
typedef __attribute__((ext_vector_type(8))) float v8f;

// ---------------------------------------------------------------------------
// zero the LN accumulators (12 doubles used; graph-replay safe re-init)
// ---------------------------------------------------------------------------
__global__ void zero_acc_k(double* a) {
    if (threadIdx.x < 16) a[threadIdx.x] = 0.0;
}

// ---------------------------------------------------------------------------
// Locally-connected conv (stride 1, VALID) + ReLU + sum/sumsq for LayerNorm.
// One wave handles 16 consecutive output pixels of one row.
// D = A(16xK weights) x B(Kx16 patches); diagonal of D = the 16 dot products.
// ---------------------------------------------------------------------------
template<int CC, int KK>
__global__ __launch_bounds__(256)
void lc_conv_wmma(const float* __restrict__ x, const float* __restrict__ w,
                  float* __restrict__ out, double* __restrict__ acc,
                  int Ho, int Wo, int Wi, long long wcs, long long xcs)
{
    constexpr int K   = KK * KK;          // per-channel patch length
    constexpr int NKB = (K + 3) >> 2;     // number of 16x16x4 WMMAs per channel
    const int lane = threadIdx.x & 31;
    const int wv   = threadIdx.x >> 5;
    const int GW   = (Wo + 15) >> 4;
    const int grp  = blockIdx.x * 8 + wv;
    if (grp >= Ho * GW) return;           // wave-uniform exit (EXEC stays all-1)
    const int i  = grp / GW;
    const int j0 = (grp - i * GW) << 4;
    const int m  = lane & 15;             // pixel index within group (A row / B col)
    const int h  = lane >> 4;             // half: k-pair select per ISA layout
    int pj = j0 + m; if (pj > Wo - 1) pj = Wo - 1;   // clamp partial groups
    const long long pixw = (long long)(i * Wo + pj) * K;

    v8f c = {};
    for (int ci = 0; ci < CC; ++ci) {
        const float* wp = w + (long long)ci * wcs + pixw;
        const float* xp = x + (long long)ci * xcs;
#pragma unroll 2
        for (int kb = 0; kb < NKB; ++kb) {
            const int k0  = (kb << 2) + (h << 1);
            const int k1  = k0 + 1;
            const int kc0 = (k0 < K) ? k0 : K - 1;   // clamp addr, zero value
            const int kc1 = (k1 < K) ? k1 : K - 1;
            // A: weights of pixel m, elements kc0/kc1 (contiguous stream per pixel)
            float a0 = wp[kc0]; a0 = (k0 < K) ? a0 : 0.f;
            float a1 = wp[kc1]; a1 = (k1 < K) ? a1 : 0.f;
            // B: patch values of pixel n (= m), elements kc0/kc1
            const int dy0 = kc0 / KK, dx0 = kc0 - dy0 * KK;
            const int dy1 = kc1 / KK, dx1 = kc1 - dy1 * KK;
            float b0 = xp[(i + dy0) * Wi + pj + dx0]; b0 = (k0 < K) ? b0 : 0.f;
            float b1 = xp[(i + dy1) * Wi + pj + dx1]; b1 = (k1 < K) ? b1 : 0.f;
            v2f av; av.x = a0; av.y = a1;
            v2f bv; bv.x = b0; bv.y = b1;
            c = __builtin_amdgcn_wmma_f32_16x16x4_f32(
                    false, av, false, bv, (short)0, c, false, false);
        }
    }

    // Diagonal extraction: D(i,i) lives at VGPR (i&7), lane (i<8 ? i : i+16).
    const int src = (lane < 8) ? lane : lane + 16;
    float res = 0.f;
#pragma unroll
    for (int r = 0; r < 8; ++r) {
        float t = __shfl(c[r], src, 32);
        if ((lane & 7) == r) res = t;
    }
    res = fmaxf(res, 0.f);                         // ReLU

    const bool valid = (lane < 16) && (j0 + lane < Wo);
    float v = valid ? res : 0.f;
    if (valid) out[i * Wo + j0 + lane] = v;

    // wave reduction of sum / sumsq, then one double atomic per wave
    float s = v, s2 = v * v;
#pragma unroll
    for (int off = 16; off; off >>= 1) {
        s  += __shfl_down(s,  off, 32);
        s2 += __shfl_down(s2, off, 32);
    }
    if (lane == 0) {
        atomicAdd(&acc[0], (double)s);
        atomicAdd(&acc[1], (double)s2);
    }
}

// ---------------------------------------------------------------------------
// LayerNorm (per-position affine) in place
// ---------------------------------------------------------------------------
__global__ __launch_bounds__(256)
void ln_ip_k(float* buf, const float* __restrict__ g, const float* __restrict__ b,
             const double* __restrict__ acc, int N)
{
    int t = blockIdx.x * 256 + threadIdx.x;
    if (t >= N) return;
    double mean = acc[0] / (double)N;
    double var  = acc[1] / (double)N - mean * mean;
    float  inv  = rsqrtf((float)var + 1e-5f);
    float  mf   = (float)mean;
    buf[t] = (buf[t] - mf) * inv * g[t] + b[t];
}

// ---------------------------------------------------------------------------
// LayerNorm + MaxPool2d(2,2) (floor), normalizing all 4 candidates (g,b vary
// per position so pooling must follow the affine)
// ---------------------------------------------------------------------------
__global__ __launch_bounds__(256)
void ln_pool_k(const float* __restrict__ in, float* __restrict__ outp,
               const float* __restrict__ g, const float* __restrict__ b,
               const double* __restrict__ acc, int Ho, int Wo)
{
    const int Hp = Ho >> 1, Wp = Wo >> 1;
    int t = blockIdx.x * 256 + threadIdx.x;
    if (t >= Hp * Wp) return;
    const int N = Ho * Wo;
    double mean = acc[0] / (double)N;
    double var  = acc[1] / (double)N - mean * mean;
    float  inv  = rsqrtf((float)var + 1e-5f);
    float  mf   = (float)mean;
    const int pi = t / Wp, pj = t - pi * Wp;
    float best = -3.4e38f;
#pragma unroll
    for (int dy = 0; dy < 2; ++dy)
#pragma unroll
        for (int dx = 0; dx < 2; ++dx) {
            int id = (2 * pi + dy) * Wo + 2 * pj + dx;
            float vv = (in[id] - mf) * inv * g[id] + b[id];
            best = fmaxf(best, vv);
        }
    outp[t] = best;
}

// ---------------------------------------------------------------------------
// Head: LN(3x3) -> flatten(9) -> FC(9x565) -> softmax(565). One block.
// ---------------------------------------------------------------------------
__global__ __launch_bounds__(1024)
void head_k(const float* __restrict__ a6, const double* __restrict__ acc,
            const float* __restrict__ g, const float* __restrict__ b,
            const float* __restrict__ fcw, const float* __restrict__ fcb,
            float* __restrict__ out)
{
    __shared__ float h[9];
    __shared__ float red[1024];
    const int t = threadIdx.x;
    if (t < 9) {
        double mean = acc[0] / 9.0;
        double var  = acc[1] / 9.0 - mean * mean;
        float  inv  = rsqrtf((float)var + 1e-5f);
        h[t] = (a6[t] - (float)mean) * inv * g[t] + b[t];
    }
    __syncthreads();
    float lg = -3.4e38f;
    if (t < 565) {
        lg = fcb[t];
#pragma unroll
        for (int r = 0; r < 9; ++r) lg += h[r] * fcw[r * 565 + t];
    }
    red[t] = lg;
    __syncthreads();
    for (int s = 512; s > 0; s >>= 1) {
        if (t < s) red[t] = fmaxf(red[t], red[t + s]);
        __syncthreads();
    }
    const float mx = red[0];
    __syncthreads();
    float e = (t < 565) ? expf(lg - mx) : 0.f;
    red[t] = e;
    __syncthreads();
    for (int s = 512; s > 0; s >>= 1) {
        if (t < s) red[t] += red[t + s];
        __syncthreads();
    }
    if (t < 565) out[t] = e / red[0];
}

// ---------------------------------------------------------------------------
// launcher
// ---------------------------------------------------------------------------
static inline int cdiv(int a, int b) { return (a + b - 1) / b; }

extern "C" void kernel_launch(void* const* d_in, const int* in_sizes, int n_in,
                              void* d_out, int out_size, void* d_ws, size_t ws_size,
                              hipStream_t stream)
{
    const float* x   = (const float*)d_in[0];
    const float* w1  = (const float*)d_in[1];
    const float* w2  = (const float*)d_in[2];
    const float* w3  = (const float*)d_in[3];
    const float* w4  = (const float*)d_in[4];
    const float* w5  = (const float*)d_in[5];
    const float* w6  = (const float*)d_in[6];
    const float* fcw = (const float*)d_in[7];
    const float* fcb = (const float*)d_in[8];
    const float* g1 = (const float*)d_in[9],  *b1 = (const float*)d_in[10];
    const float* g2 = (const float*)d_in[11], *b2 = (const float*)d_in[12];
    const float* g3 = (const float*)d_in[13], *b3 = (const float*)d_in[14];
    const float* g4 = (const float*)d_in[15], *b4 = (const float*)d_in[16];
    const float* g5 = (const float*)d_in[17], *b5 = (const float*)d_in[18];
    const float* g6 = (const float*)d_in[19], *b6 = (const float*)d_in[20];
    float* outp = (float*)d_out;

    // workspace layout: 16 doubles (LN accumulators, 2 per layer) then floats
    double* acc = (double*)d_ws;
    float*  fb  = (float*)((char*)d_ws + 256);
    float* B1 = fb;            // 794*794 conv1 out
    float* P1 = B1 + 630436;   // 397*397 pooled
    float* B2 = P1 + 157609;   // 374*374 conv2 out (LN in place)
    float* B3 = B2 + 139876;   // 348*348 conv3 out
    float* P3 = B3 + 121104;   // 174*174 pooled
    float* B4 = P3 + 30276;    // 148*148 conv4 out (LN in place)
    float* B5 = B4 + 21904;    // 101*101 conv5 out
    float* P5 = B5 + 10201;    // 50*50 pooled
    float* B6 = P5 + 2500;     // 3*3 conv6 out

    zero_acc_k<<<1, 32, 0, stream>>>(acc);

    // layer 1: C=3, k=7, 800 -> 794, then LN + pool -> 397
    {
        int Ho = 794, Wo = 794, Wi = 800;
        int groups = Ho * cdiv(Wo, 16);
        lc_conv_wmma<3, 7><<<cdiv(groups, 8), 256, 0, stream>>>(
            x, w1, B1, acc + 0, Ho, Wo, Wi, 30891364LL, 640000LL);
        ln_pool_k<<<cdiv(397 * 397, 256), 256, 0, stream>>>(B1, P1, g1, b1, acc + 0, Ho, Wo);
    }
    // layer 2: k=24, 397 -> 374, LN in place
    {
        int Ho = 374, Wo = 374, Wi = 397;
        int groups = Ho * cdiv(Wo, 16);
        lc_conv_wmma<1, 24><<<cdiv(groups, 8), 256, 0, stream>>>(
            P1, w2, B2, acc + 2, Ho, Wo, Wi, 0LL, 0LL);
        ln_ip_k<<<cdiv(374 * 374, 256), 256, 0, stream>>>(B2, g2, b2, acc + 2, 374 * 374);
    }
    // layer 3: k=27, 374 -> 348, LN + pool -> 174
    {
        int Ho = 348, Wo = 348, Wi = 374;
        int groups = Ho * cdiv(Wo, 16);
        lc_conv_wmma<1, 27><<<cdiv(groups, 8), 256, 0, stream>>>(
            B2, w3, B3, acc + 4, Ho, Wo, Wi, 0LL, 0LL);
        ln_pool_k<<<cdiv(174 * 174, 256), 256, 0, stream>>>(B3, P3, g3, b3, acc + 4, Ho, Wo);
    }
    // layer 4: k=27, 174 -> 148, LN in place
    {
        int Ho = 148, Wo = 148, Wi = 174;
        int groups = Ho * cdiv(Wo, 16);
        lc_conv_wmma<1, 27><<<cdiv(groups, 8), 256, 0, stream>>>(
            P3, w4, B4, acc + 6, Ho, Wo, Wi, 0LL, 0LL);
        ln_ip_k<<<cdiv(148 * 148, 256), 256, 0, stream>>>(B4, g4, b4, acc + 6, 148 * 148);
    }
    // layer 5: k=48, 148 -> 101, LN + pool -> 50
    {
        int Ho = 101, Wo = 101, Wi = 148;
        int groups = Ho * cdiv(Wo, 16);
        lc_conv_wmma<1, 48><<<cdiv(groups, 8), 256, 0, stream>>>(
            B4, w5, B5, acc + 8, Ho, Wo, Wi, 0LL, 0LL);
        ln_pool_k<<<cdiv(50 * 50, 256), 256, 0, stream>>>(B5, P5, g5, b5, acc + 8, Ho, Wo);
    }
    // layer 6: k=48, 50 -> 3
    {
        int Ho = 3, Wo = 3, Wi = 50;
        int groups = Ho * cdiv(Wo, 16);
        lc_conv_wmma<1, 48><<<cdiv(groups, 8), 256, 0, stream>>>(
            P5, w6, B6, acc + 10, Ho, Wo, Wi, 0LL, 0LL);
    }
    // head: LN(9) + FC + softmax
    head_k<<<1, 1024, 0, stream>>>(B6, acc + 10, g6, b6, fcw, fcb, outp);
}